// LocalGNNCliqueLine_16217796509772
// MI455X (gfx1250) — compile-verified
//
#include <hip/hip_runtime.h>
#include <hip/hip_bf16.h>

typedef float v2f __attribute__((ext_vector_type(2)));
typedef float v8f __attribute__((ext_vector_type(8)));

// ---------------------------------------------------------------------------
// Z[Mrows x N] = H[Mrows x N] @ S[N x N], all row-major fp32.
// One wave owns NT adjacent 16x16 output tiles (shared A fragment), K-loop in
// steps of 4 using V_WMMA_F32_16X16X4_F32.
//   A tile (16x4):  lane = m + 16*(k>=2); a.x = H[m, k+2*hi], a.y = +1
//   B tile (4x16):  VGPR v: lanes 0-15 -> K=v, lanes 16-31 -> K=v+2
//   C/D (16x16):    VGPR v: lanes 0-15 -> M=v, lanes 16-31 -> M=v+8
// N is compile-time so all strides fold into immediate offsets; FULLM removes
// the partial-tile masking for Mrows % 16 == 0.
// ---------------------------------------------------------------------------
template <int N, int NT, bool FULLM>
__global__ __launch_bounds__(256) void gemm_wmma(
    float* __restrict__ Z, const float* __restrict__ H,
    const float* __restrict__ S, int Mrows)
{
    const int lane    = threadIdx.x & 31;
    const int wave    = (blockIdx.x * blockDim.x + threadIdx.x) >> 5;
    const int nStrips = N / (16 * NT);
    const int mTiles  = (Mrows + 15) >> 4;
    if (wave >= mTiles * nStrips) return;          // wave-uniform

    const int mTile  = wave / nStrips;
    const int nStrip = wave - mTile * nStrips;
    const int m0 = mTile * 16;
    const int n0 = nStrip * (16 * NT);

    const int col  = lane & 15;    // 0..15
    const int hi   = lane >> 4;    // 0 or 1
    const int koff = hi << 1;      // 0 or 2
    const int m    = m0 + col;

    const float scale = (m < Mrows) ? 1.0f : 0.0f;
    const float* __restrict__ Hrow =
        H + (size_t)(FULLM ? m : (m < Mrows ? m : 0)) * N + koff;
    const float* __restrict__ Sbase = S + (size_t)koff * N + n0 + col;

    v8f acc[NT];
#pragma unroll
    for (int t = 0; t < NT; ++t)
        acc[t] = (v8f){0.f, 0.f, 0.f, 0.f, 0.f, 0.f, 0.f, 0.f};

#pragma unroll 8
    for (int k = 0; k < N; k += 4) {
        v2f a;
        a.x = Hrow[k];        // H[m, k + koff]
        a.y = Hrow[k + 1];    // H[m, k + koff + 1]
        if (!FULLM) { a.x *= scale; a.y *= scale; }

        const float* __restrict__ Sp = Sbase + (size_t)k * N;
#pragma unroll
        for (int t = 0; t < NT; ++t) {
            v2f b;
            b.x = Sp[t * 16];        // S[k + koff,     n0 + 16t + col]
            b.y = Sp[t * 16 + N];    // S[k + koff + 1, n0 + 16t + col]
            acc[t] = __builtin_amdgcn_wmma_f32_16x16x4_f32(
                false, a, false, b, (short)0, acc[t], false, false);
        }
    }

#pragma unroll
    for (int t = 0; t < NT; ++t) {
#pragma unroll
        for (int v = 0; v < 8; ++v) {
            const int mrow = m0 + v + (hi << 3);
            if (FULLM || mrow < Mrows)
                Z[(size_t)mrow * N + n0 + t * 16 + col] = acc[t][v];
        }
    }
}

// ---------------------------------------------------------------------------
// Hout[b,g,n] = relu( sum_f  W[0,f,g]*Z0[b,f,n] + W[1,f,g]*Z1[b,f,n]
//                          + W[2,f,g]*Z2[b,f,n]  + bias[g] )
// ---------------------------------------------------------------------------
__global__ void combine_relu_kernel(
    float* __restrict__ Hout,
    const float* __restrict__ Z0, const float* __restrict__ Z1,
    const float* __restrict__ Z2,
    const float* __restrict__ W, const float* __restrict__ bias,
    int Bn, int Fin, int Fout, int N)
{
    const int tid = blockIdx.x * blockDim.x + threadIdx.x;
    if (tid >= Bn * N) return;
    const int n = tid % N;
    const int b = tid / N;

    float out[16];
    for (int g = 0; g < Fout; ++g) out[g] = bias[g];

    for (int f = 0; f < Fin; ++f) {
        const size_t idx = ((size_t)(b * Fin + f)) * (size_t)N + n;
        const float z0 = Z0[idx];
        const float z1 = Z1[idx];
        const float z2 = Z2[idx];
        const float* __restrict__ W0 = W + ((size_t)(0 * Fin + f)) * Fout;
        const float* __restrict__ W1 = W + ((size_t)(1 * Fin + f)) * Fout;
        const float* __restrict__ W2 = W + ((size_t)(2 * Fin + f)) * Fout;
        for (int g = 0; g < Fout; ++g)
            out[g] = fmaf(z0, W0[g], fmaf(z1, W1[g], fmaf(z2, W2[g], out[g])));
    }

    for (int g = 0; g < Fout; ++g) {
        const float v = out[g];
        Hout[((size_t)(b * Fout + g)) * (size_t)N + n] = v > 0.f ? v : 0.f;
    }
}

// ---------------------------------------------------------------------------
// P[b,f,m] = max(0, max_d Hin[b,f, incid[m,d]])   (clique -> line pooling)
// ---------------------------------------------------------------------------
__global__ void pool_kernel(
    float* __restrict__ P, const float* __restrict__ Hin,
    const int* __restrict__ incid, int BF, int N1, int N2, int DEG)
{
    const int tid = blockIdx.x * blockDim.x + threadIdx.x;
    if (tid >= BF * N2) return;
    const int m  = tid % N2;
    const int bf = tid / N2;
    const float* __restrict__ row = Hin + (size_t)bf * (size_t)N1;
    float v = 0.0f;
    for (int d = 0; d < DEG; ++d) {
        const int j = incid[m * DEG + d];
        v = fmaxf(v, row[j]);
    }
    P[tid] = v;
}

// ---------------------------------------------------------------------------
// Per-node readout MLP: y0 = relu(h^T @ R0W + R0b), y1 = y0 @ R1W + R1b.
// out is (B, 2, N2).
// ---------------------------------------------------------------------------
__global__ void readout_kernel(
    float* __restrict__ Y, const float* __restrict__ H,
    const float* __restrict__ R0W, const float* __restrict__ R0b,
    const float* __restrict__ R1W, const float* __restrict__ R1b,
    int Bn, int N2)
{
    const int tid = blockIdx.x * blockDim.x + threadIdx.x;
    if (tid >= Bn * N2) return;
    const int n = tid % N2;
    const int b = tid / N2;

    float v[16];
    for (int i = 0; i < 16; ++i)
        v[i] = H[((size_t)(b * 16 + i)) * (size_t)N2 + n];

    float y0[16];
    for (int j = 0; j < 16; ++j) {
        float s = R0b[j];
        for (int i = 0; i < 16; ++i) s = fmaf(v[i], R0W[i * 16 + j], s);
        y0[j] = s > 0.f ? s : 0.f;
    }

    for (int r = 0; r < 2; ++r) {
        float s = R1b[r];
        for (int j = 0; j < 16; ++j) s = fmaf(y0[j], R1W[j * 2 + r], s);
        Y[((size_t)(b * 2 + r)) * (size_t)N2 + n] = s;
    }
}

// ---------------------------------------------------------------------------
extern "C" void kernel_launch(void* const* d_in, const int* in_sizes, int n_in,
                              void* d_out, int out_size, void* d_ws, size_t ws_size,
                              hipStream_t stream)
{
    (void)in_sizes; (void)n_in; (void)out_size; (void)ws_size;

    const float* x     = (const float*)d_in[0];   // (4,1,8192)
    const float* S1    = (const float*)d_in[1];   // (8192,8192)
    const float* S2    = (const float*)d_in[2];   // (4096,4096)
    const int*   incid = (const int*)  d_in[3];   // (4096,4)
    const float* W10   = (const float*)d_in[4];   // (3,1,16)
    const float* b10   = (const float*)d_in[5];
    const float* W11   = (const float*)d_in[6];   // (3,16,16)
    const float* b11   = (const float*)d_in[7];
    const float* W20   = (const float*)d_in[8];   // (3,16,16)
    const float* b20   = (const float*)d_in[9];
    const float* R0W   = (const float*)d_in[10];  // (16,16)
    const float* R0b   = (const float*)d_in[11];
    const float* R1W   = (const float*)d_in[12];  // (16,2)
    const float* R1b   = (const float*)d_in[13];
    float* out = (float*)d_out;

    const int B = 4, F1 = 16, N1 = 8192, N2 = 4096, DEG = 4;

    // Workspace layout (total ~9 MB)
    char* ws = (char*)d_ws;
    const size_t szBig = (size_t)(B * F1) * N1 * sizeof(float); // 2 MB
    float* h1 = (float*)(ws);                 // layer-1 output, reused as h3
    float* tA = (float*)(ws + 1 * szBig);     // z1 / t1 / u1
    float* tB = (float*)(ws + 2 * szBig);     // z2 / t2 / u2
    float* h2 = (float*)(ws + 3 * szBig);     // layer-2 output
    float* p  = (float*)(ws + 4 * szBig);     // pooled (B,F1,N2), 1 MB

    auto blocksFor = [](int tiles) { return (tiles + 7) / 8; }; // 8 waves/block
    auto grid1d    = [](int total, int blk) { return (total + blk - 1) / blk; };

    // ---- Layer 1: zs = [x, x@S1, x@S1@S1], fold with W10 ----
    // M=4 GEMMs: NT=1, partial tile.
    {
        const int tiles = 1 * (N1 / 16);
        gemm_wmma<N1, 1, false><<<blocksFor(tiles), 256, 0, stream>>>(tA, x,  S1, B);
        gemm_wmma<N1, 1, false><<<blocksFor(tiles), 256, 0, stream>>>(tB, tA, S1, B);
    }
    combine_relu_kernel<<<grid1d(B * N1, 256), 256, 0, stream>>>(
        h1, x, tA, tB, W10, b10, B, 1, F1, N1);

    // ---- Layer 2: zs = [h1, h1@S1, h1@S1@S1], fold with W11 ----
    // M=64 GEMMs: NT=2, full tiles.
    {
        const int tiles = 4 * (N1 / 32);
        gemm_wmma<N1, 2, true><<<blocksFor(tiles), 256, 0, stream>>>(tA, h1, S1, B * F1);
        gemm_wmma<N1, 2, true><<<blocksFor(tiles), 256, 0, stream>>>(tB, tA, S1, B * F1);
    }
    combine_relu_kernel<<<grid1d(B * N1, 256), 256, 0, stream>>>(
        h2, h1, tA, tB, W11, b11, B, F1, F1, N1);

    // ---- Pool clique -> line ----
    pool_kernel<<<grid1d(B * F1 * N2, 256), 256, 0, stream>>>(
        p, h2, incid, B * F1, N1, N2, DEG);

    // ---- Layer 3 on line graph: zs = [p, p@S2, p@S2@S2], fold with W20 ----
    {
        const int tiles = 4 * (N2 / 32);
        gemm_wmma<N2, 2, true><<<blocksFor(tiles), 256, 0, stream>>>(tA, p,  S2, B * F1);
        gemm_wmma<N2, 2, true><<<blocksFor(tiles), 256, 0, stream>>>(tB, tA, S2, B * F1);
    }
    combine_relu_kernel<<<grid1d(B * N2, 256), 256, 0, stream>>>(
        h1 /*reuse as h3*/, p, tA, tB, W20, b20, B, F1, F1, N2);

    // ---- Readout MLP ----
    readout_kernel<<<grid1d(B * N2, 256), 256, 0, stream>>>(
        out, h1, R0W, R0b, R1W, R1b, B, N2);
}